// SparseConvNetBackbone_51007031608109
// MI455X (gfx1250) — compile-verified
//
#include <hip/hip_runtime.h>

// Receptive-field-pruned SparseConvNet backbone for MI455X (gfx1250, wave32).
// Only the center voxel x[:,:,8,8,8] of the final 16^3 map feeds the head, so the
// exact computation collapses to:
//   voxel[28..36]^3 -> x1[32][7^3] -> x2[64][3^3] -> x3[128] -> out[256]   (per batch)
// ~75 MFLOP / ~1.5 MB weight traffic total => latency-bound; we minimize kernel
// count, prefetch one-shot weights, use the TDM for LDS staging, and run the two
// GEMM-shaped stages on the fp32 matrix pipe (v_wmma_f32_16x16x4_f32, exact fp32).

typedef __attribute__((ext_vector_type(2))) float v2f;
typedef __attribute__((ext_vector_type(8))) float v8f;

#define RES 64
#define BN_EPS 1e-5f

// ---------------------------------------------------------------------------
// TDM helper: 1-D tensor_load_to_lds of `n4` fp32 elements (global -> LDS).
// D# per CDNA5 ISA §8.3/8.4: group0 = {count=1, lds_addr, global_addr, type=2},
// group1 = {data_size=4B, tensor_dim0=n4, tile_dim0=n4}, groups 2/3 zero.
// ---------------------------------------------------------------------------
#if defined(__has_builtin)
#if __has_builtin(__builtin_amdgcn_tensor_load_to_lds) && \
    __has_builtin(__builtin_amdgcn_s_wait_tensorcnt)
#define USE_TDM 1
#endif
#endif

#if defined(USE_TDM)
typedef __attribute__((ext_vector_type(4))) unsigned int u32x4;
typedef __attribute__((ext_vector_type(4))) int          i32x4;
typedef __attribute__((ext_vector_type(8))) int          i32x8;

__device__ __forceinline__ void tdm_load_1d(unsigned lds_addr, const void* gsrc,
                                            unsigned n4) {
    unsigned long long ga = (unsigned long long)(size_t)gsrc;
    u32x4 g0 = { 1u,                                   // count=1, user desc
                 lds_addr,                             // LDS byte address
                 (unsigned)ga,                         // global_addr[31:0]
                 (unsigned)(ga >> 32) | (2u << 30) };  // addr[56:32] | type=2
    i32x8 g1 = { (int)(2u << 16),                      // wg_mask=0, data_size=4B
                 (int)((n4 & 0xffffu) << 16),          // tensor_dim0[15:0]
                 (int)(n4 >> 16),                      // tensor_dim0[31:16]
                 (int)(n4 << 16),                      // tile_dim0
                 0, 0, 0, 0 };
    i32x4 z4 = { 0, 0, 0, 0 };
#if __clang_major__ >= 23
    i32x8 z8 = { 0, 0, 0, 0, 0, 0, 0, 0 };
    __builtin_amdgcn_tensor_load_to_lds(g0, g1, z4, z4, z8, 0);
#else
    __builtin_amdgcn_tensor_load_to_lds(g0, g1, z4, z4, 0);
#endif
}
#endif

// ---------------------------------------------------------------------------
// Kernel 1: 2 blocks per batch (block = (batch, c2-half)). Both halves compute
// the tiny conv1 redundantly in LDS, then each produces 32 of the 64 c2 rows of
// x2 at positions {15,16,17}^3, stored as B-matrix x2ws[k=c2*27+tap][batch].
// ---------------------------------------------------------------------------
__global__ __launch_bounds__(256) void conv12_kernel(
    const float* __restrict__ voxel, const float* __restrict__ w1,
    const float* __restrict__ g1, const float* __restrict__ b1,
    const float* __restrict__ m1, const float* __restrict__ v1,
    const float* __restrict__ w2,
    const float* __restrict__ g2, const float* __restrict__ b2,
    const float* __restrict__ m2, const float* __restrict__ v2,
    float* __restrict__ x2ws)
{
    __shared__ float vox[9 * 9 * 9];      // voxel[28..36]^3
    __shared__ float sw1[32 * 27];
    __shared__ float x1[32 * 343];        // x1[c1][7^3], coords 29..35
    __shared__ float sc1[32], sh1[32];
    __shared__ float sc2[64], sh2[64];

    const int tid    = threadIdx.x;
    const int bb     = blockIdx.x;        // 32 blocks
    const int b      = bb >> 1;           // batch
    const int c2base = (bb & 1) * 32;     // this block's c2 half
    const float* vbase = voxel + (size_t)b * RES * RES * RES;
    const float* w2h   = w2 + (size_t)c2base * (32 * 27);

    // Warm this block's 110 KB w2 half into L2/L0 while conv1 runs (128B stride).
    for (int i = tid * 32; i < 32 * 32 * 27; i += 256 * 32)
        __builtin_prefetch(w2h + i, 0, 1);

    for (int i = tid; i < 729; i += 256) {
        int d = i / 81, h = (i / 9) % 9, w = i % 9;
        vox[i] = vbase[(size_t)(d + 28) * RES * RES + (h + 28) * RES + (w + 28)];
    }
    for (int i = tid; i < 32 * 27; i += 256) sw1[i] = w1[i];
    if (tid < 32) {
        float inv = rsqrtf(v1[tid] + BN_EPS);
        float s = g1[tid] * inv;
        sc1[tid] = s; sh1[tid] = b1[tid] - m1[tid] * s;
    }
    if (tid >= 32 && tid < 96) {
        int c = tid - 32;
        float inv = rsqrtf(v2[c] + BN_EPS);
        float s = g2[c] * inv;
        sc2[c] = s; sh2[c] = b2[c] - m2[c] * s;
    }
    __syncthreads();

    // conv1 (stride 1, pad 1, interior): x1 coord 29+p reads voxel 28+p+k.
    for (int i = tid; i < 32 * 343; i += 256) {
        int c = i / 343, pos = i % 343;
        int pd = pos / 49, ph = (pos / 7) % 7, pw = pos % 7;
        const float* wc = sw1 + c * 27;
        float s = 0.f;
        #pragma unroll
        for (int kd = 0; kd < 3; ++kd)
            #pragma unroll
            for (int kh = 0; kh < 3; ++kh)
                #pragma unroll
                for (int kw = 0; kw < 3; ++kw)
                    s += vox[(pd + kd) * 81 + (ph + kh) * 9 + (pw + kw)] *
                         wc[kd * 9 + kh * 3 + kw];
        s = s * sc1[c] + sh1[c];
        x1[i] = s > 0.f ? s : 0.f;
    }
    __syncthreads();

    // conv2 (stride 2, pad 1): x2 coord 15+q reads x1 coord 29 + (2q+k).
    for (int i = tid; i < 32 * 27; i += 256) {
        int c2 = c2base + i / 27, q = i % 27;
        int qd = q / 9, qh = (q / 3) % 3, qw = q % 3;
        const float* w2r = w2 + (size_t)c2 * (32 * 27);
        float s = 0.f;
        for (int c1 = 0; c1 < 32; ++c1) {
            const float* x1c = x1 + c1 * 343;
            const float* wk  = w2r + c1 * 27;
            #pragma unroll
            for (int kd = 0; kd < 3; ++kd)
                #pragma unroll
                for (int kh = 0; kh < 3; ++kh) {
                    int base = (2 * qd + kd) * 49 + (2 * qh + kh) * 7 + 2 * qw;
                    int wb   = kd * 9 + kh * 3;
                    s += x1c[base + 0] * wk[wb + 0]
                       + x1c[base + 1] * wk[wb + 1]
                       + x1c[base + 2] * wk[wb + 2];
                }
        }
        s = s * sc2[c2] + sh2[c2];
        s = s > 0.f ? s : 0.f;
        x2ws[(size_t)(c2 * 27 + q) * 16 + b] = s;
    }
}

// ---------------------------------------------------------------------------
// Kernel 2: single workgroup, 8 waves.
//  Phase 1: D[128,16] = W3[128,1728] x X2col[1728,16] via v_wmma_f32_16x16x4_f32
//           (wave w owns c3 rows [16w,16w+16); B chunks TDM-staged into LDS).
//  Phase 2: Out[16,256] = X3^T[16,128] x Wp[128,256] + bp (wave w owns 2 N-tiles).
// A 16x4 f32 layout: lane m=l&15; VGPR0=K(2*half), VGPR1=K(2*half+1).
// B 4x16 layout mirrored; C/D 16x16: VGPR r -> M = r + 8*half, N = l&15.
// ---------------------------------------------------------------------------
#define KTOT 1728
#define KCHUNK 432

__global__ __launch_bounds__(256) void conv3_proj_kernel(
    const float* __restrict__ w3,
    const float* __restrict__ g3, const float* __restrict__ b3,
    const float* __restrict__ m3, const float* __restrict__ v3,
    const float* __restrict__ wp, const float* __restrict__ bp,
    const float* __restrict__ x2ws, float* __restrict__ out)
{
    __shared__ float sB[KCHUNK * 16];   // 27.6 KB chunk of X2col
    __shared__ float sX3[128 * 16];     // x3 as [c3][batch]
    __shared__ float sc3[128], sh3[128];

    const int tid  = threadIdx.x;
    const int lane = tid & 31;
    const int wave = tid >> 5;          // 0..7
    const int half = lane >> 4;         // 0/1
    const int l16  = lane & 15;

    // Warm wp (128 KB) for phase 2 and this wave's first w3 A-panel.
    for (int i = tid * 32; i < 128 * 256; i += 256 * 32)
        __builtin_prefetch(wp + i, 0, 1);
    const float* a_row = w3 + (size_t)(16 * wave + l16) * KTOT;
    for (int i = 0; i < KCHUNK; i += 32)
        __builtin_prefetch(a_row + i, 0, 1);

    if (tid < 128) {
        float inv = rsqrtf(v3[tid] + BN_EPS);
        float s = g3[tid] * inv;
        sc3[tid] = s; sh3[tid] = b3[tid] - m3[tid] * s;
    }

    // ---- Phase 1: conv3 GEMM, M=128 N=16 K=1728 ----
    v8f acc = {};
    for (int kc = 0; kc < KTOT; kc += KCHUNK) {
        __syncthreads();
#if defined(USE_TDM)
        if (wave == 0) {
            tdm_load_1d((unsigned)(size_t)&sB[0], x2ws + (size_t)kc * 16,
                        KCHUNK * 16);
            __builtin_amdgcn_s_wait_tensorcnt(0);
        }
#else
        for (int i = tid; i < KCHUNK * 16; i += 256)
            sB[i] = x2ws[(size_t)kc * 16 + i];
#endif
        __syncthreads();
        for (int kl = 0; kl < KCHUNK; kl += 4) {
            v2f a = *(const v2f*)(a_row + kc + kl + 2 * half);
            v2f bv;
            bv[0] = sB[(kl + 2 * half) * 16 + l16];
            bv[1] = sB[(kl + 2 * half + 1) * 16 + l16];
            acc = __builtin_amdgcn_wmma_f32_16x16x4_f32(
                false, a, false, bv, (short)0, acc, false, false);
        }
    }

    // BN3 + ReLU; scatter x3 into LDS as [c3][batch].
    #pragma unroll
    for (int r = 0; r < 8; ++r) {
        int c3 = 16 * wave + r + 8 * half;
        float val = acc[r] * sc3[c3] + sh3[c3];
        val = val > 0.f ? val : 0.f;
        sX3[c3 * 16 + l16] = val;
    }
    __syncthreads();

    // ---- Phase 2: projection GEMM, M=16(batch) N=256 K=128 ----
    for (int t = 0; t < 2; ++t) {
        const int nb = (2 * wave + t) * 16;
        v8f acc2 = {};
        for (int k = 0; k < 128; k += 4) {
            v2f a;
            a[0] = sX3[(k + 2 * half) * 16 + l16];       // A[m=batch][k']
            a[1] = sX3[(k + 2 * half + 1) * 16 + l16];
            v2f bv;
            bv[0] = wp[(size_t)(k + 2 * half) * 256 + nb + l16];
            bv[1] = wp[(size_t)(k + 2 * half + 1) * 256 + nb + l16];
            acc2 = __builtin_amdgcn_wmma_f32_16x16x4_f32(
                false, a, false, bv, (short)0, acc2, false, false);
        }
        float bias = bp[nb + l16];
        #pragma unroll
        for (int r = 0; r < 8; ++r) {
            int batch = r + 8 * half;
            out[batch * 256 + nb + l16] = acc2[r] + bias;
        }
    }
}

// ---------------------------------------------------------------------------
extern "C" void kernel_launch(void* const* d_in, const int* in_sizes, int n_in,
                              void* d_out, int out_size, void* d_ws, size_t ws_size,
                              hipStream_t stream) {
    const float* voxel = (const float*)d_in[0];
    const float* w1 = (const float*)d_in[1];
    const float* g1 = (const float*)d_in[2];
    const float* b1 = (const float*)d_in[3];
    const float* m1 = (const float*)d_in[4];
    const float* v1 = (const float*)d_in[5];
    const float* w2 = (const float*)d_in[6];
    const float* g2 = (const float*)d_in[7];
    const float* b2 = (const float*)d_in[8];
    const float* m2 = (const float*)d_in[9];
    const float* v2 = (const float*)d_in[10];
    const float* w3 = (const float*)d_in[11];
    const float* g3 = (const float*)d_in[12];
    const float* b3 = (const float*)d_in[13];
    const float* m3 = (const float*)d_in[14];
    const float* v3 = (const float*)d_in[15];
    const float* wp = (const float*)d_in[16];
    const float* bp = (const float*)d_in[17];

    float* x2ws = (float*)d_ws;   // 1728 * 16 * 4 = 110,592 bytes

    conv12_kernel<<<32, 256, 0, stream>>>(voxel, w1, g1, b1, m1, v1,
                                          w2, g2, b2, m2, v2, x2ws);
    conv3_proj_kernel<<<1, 256, 0, stream>>>(w3, g3, b3, m3, v3,
                                             wp, bp, x2ws, (float*)d_out);
}